// GENEO_6975026889201
// MI455X (gfx1250) — compile-verified
//
#include <hip/hip_runtime.h>

#define LIMG   1024
#define TILE   32
#define HALO   19
#define LDSW   70                     // row stride (floats) == window width (TDM fills contiguously)
#define LDSH   (TILE + 2*HALO)        // 70
#define LDSE   (LDSH * LDSW)         // 4900
#define NTHR   512
#define BIGV   1e9f

typedef unsigned int u32x4 __attribute__((ext_vector_type(4)));
typedef int          i32x8 __attribute__((ext_vector_type(8)));
typedef int          i32x4 __attribute__((ext_vector_type(4)));

// ---- compile-time ring offset tables (exact integer reformulation of
// |sqrt(dy^2+dx^2)-r| <= 0.5  <=>  (2r-1)^2 <= 4(dy^2+dx^2) <= (2r+1)^2;
// no lattice point lies on the boundary, so this matches the float reference) ----
constexpr int ringCount(int r) {
  int c = 0;
  for (int dy = -r; dy <= r; ++dy)
    for (int dx = -r; dx <= r; ++dx) {
      int s4 = 4 * (dy * dy + dx * dx);
      if (s4 >= (2 * r - 1) * (2 * r - 1) && s4 <= (2 * r + 1) * (2 * r + 1)) ++c;
    }
  return c;
}

template <int R> struct RingTab { short o[ringCount(R)]; };

template <int R> constexpr RingTab<R> makeRing() {
  RingTab<R> t{};
  int n = 0;
  for (int dy = -R; dy <= R; ++dy)
    for (int dx = -R; dx <= R; ++dx) {
      int s4 = 4 * (dy * dy + dx * dx);
      if (s4 >= (2 * R - 1) * (2 * R - 1) && s4 <= (2 * R + 1) * (2 * R + 1))
        t.o[n++] = (short)((dy + HALO) * LDSW + (dx + HALO));   // in [0, 2698]
    }
  return t;
}

constexpr auto T7  = makeRing<7>();
constexpr auto T11 = makeRing<11>();
constexpr auto T15 = makeRing<15>();
constexpr auto T19 = makeRing<19>();

__global__ __launch_bounds__(NTHR)
void geneo_layer(const float* __restrict__ src, float* __restrict__ dst,
                 const float* __restrict__ P1, const float* __restrict__ P2,
                 const float* __restrict__ P3, const float* __restrict__ P4)
{
  __shared__ float tile[LDSE];

  const int t  = threadIdx.x;
  const int ox = blockIdx.x * TILE;
  const int oy = blockIdx.y * TILE;

  // Interior blocks: the whole 70x70 window is inside the image -> one TDM tile load.
  const bool interior = (ox >= HALO) && (oy >= HALO) &&
                        (ox + TILE + HALO <= LIMG) && (oy + TILE + HALO <= LIMG);

#if __has_builtin(__builtin_amdgcn_tensor_load_to_lds)
  if (interior) {
    if (t == 0) {                           // TDM ignores EXEC; issue from one wave, one lane path
      const float* gp = src + ((size_t)(oy - HALO) * LIMG + (ox - HALO));
      unsigned long long ga = (unsigned long long)(uintptr_t)gp;
      unsigned ldsb = (unsigned)(uintptr_t)&tile[0];      // low 32 bits = LDS byte offset

      // D# group 0: count=1 | lds_addr | global_addr[56:0] | type=2 (ISA 8.3)
      u32x4 g0 = { 1u,
                   ldsb,
                   (unsigned)(ga & 0xFFFFFFFFu),
                   (unsigned)((ga >> 32) & 0x01FFFFFFu) | (2u << 30) };
      // D# group 1: data_size=4B; tensor_dim0=70, tensor_dim1=70;
      // tile_dim0=70, tile_dim1=70, tile_dim2=0; tensor_dim0_stride=1024 (ISA 8.4)
      i32x8 g1 = { (int)(2u << 16),          // [17:16] data_size=2 (4 bytes); mask/flags=0
                   (int)(70u << 16),         // [31:16] tensor_dim0 lo16
                   (int)(70u << 16),         // [15:0] dim0 hi=0 | [31:16] tensor_dim1 lo16
                   (int)(70u << 16),         // [15:0] dim1 hi=0 | [31:16] tile_dim0=70
                   (int)(70u),               // [15:0] tile_dim1=70 | [31:16] tile_dim2=0
                   (int)(LIMG),              // tensor_dim0_stride lo32 = 1024
                   0, 0 };                   // stride hi / tensor_dim1_stride = 0
      i32x4 gz4 = { 0, 0, 0, 0 };
      i32x8 gz8 = { 0, 0, 0, 0, 0, 0, 0, 0 };
      // clang-23 / therock form: 6 args (g0, g1, g2, g3, extra, cpol)
      __builtin_amdgcn_tensor_load_to_lds(g0, g1, gz4, gz4, gz8, 0);
      __builtin_amdgcn_s_wait_tensorcnt(0);
    }
  } else
#endif
  {
    // Boundary (or fallback) path: async global->LDS for in-bounds, BIG pad for OOB.
    #pragma unroll
    for (int i = 0; i < (LDSE + NTHR - 1) / NTHR; ++i) {
      int e = t + i * NTHR;
      if (e < LDSE) {
        int ly = e / LDSW, lx = e - ly * LDSW;
        int gy = oy + ly - HALO, gx = ox + lx - HALO;
        if (((unsigned)gy < LIMG) && ((unsigned)gx < LIMG)) {
          unsigned voff = (unsigned)((gy * LIMG + gx) * 4);
          unsigned lad  = (unsigned)(uintptr_t)&tile[e];
          asm volatile("global_load_async_to_lds_b32 %0, %1, %2"
                       :: "v"(lad), "v"(voff), "s"(src) : "memory");
        } else {
          tile[e] = BIGV;
        }
      }
    }
    asm volatile("s_wait_asynccnt 0" ::: "memory");
  }
  __syncthreads();

  // ---- compute: each thread owns 2 pixels: (tw, tx) and (tw+16, tx) ----
  const int tx = t & (TILE - 1);
  const int tw = t >> 5;                 // 0..15 (one wave == one tile row -> conflict-free)
  const int b0 = tw * LDSW + tx;         // tile[b0 + ringoff] == neighbor (offsets all >= 0)
  const int RB = 16 * LDSW;              // second-pixel row fold (compile-time immediate)
  const int CC = HALO * LDSW + HALO;     // center offset

  const float c0 = tile[b0 + CC];
  const float c1 = tile[b0 + RB + CC];

  float m7a = BIGV, m7b = BIGV, m11a = BIGV, m11b = BIGV;
  float m15a = BIGV, m15b = BIGV, m19a = BIGV, m19b = BIGV;

  // all ring counts are even; pairing lets the backend form v_min3 (min is exact/associative)
  #pragma unroll
  for (int i = 0; i < ringCount(7); i += 2) {
    const int o0 = T7.o[i], o1 = T7.o[i + 1];
    m7a = fminf(m7a, fminf(fabsf(c0 - tile[b0 + o0]),      fabsf(c0 - tile[b0 + o1])));
    m7b = fminf(m7b, fminf(fabsf(c1 - tile[b0 + RB + o0]), fabsf(c1 - tile[b0 + RB + o1])));
  }
  #pragma unroll
  for (int i = 0; i < ringCount(11); i += 2) {
    const int o0 = T11.o[i], o1 = T11.o[i + 1];
    m11a = fminf(m11a, fminf(fabsf(c0 - tile[b0 + o0]),      fabsf(c0 - tile[b0 + o1])));
    m11b = fminf(m11b, fminf(fabsf(c1 - tile[b0 + RB + o0]), fabsf(c1 - tile[b0 + RB + o1])));
  }
  #pragma unroll
  for (int i = 0; i < ringCount(15); i += 2) {
    const int o0 = T15.o[i], o1 = T15.o[i + 1];
    m15a = fminf(m15a, fminf(fabsf(c0 - tile[b0 + o0]),      fabsf(c0 - tile[b0 + o1])));
    m15b = fminf(m15b, fminf(fabsf(c1 - tile[b0 + RB + o0]), fabsf(c1 - tile[b0 + RB + o1])));
  }
  #pragma unroll
  for (int i = 0; i < ringCount(19); i += 2) {
    const int o0 = T19.o[i], o1 = T19.o[i + 1];
    m19a = fminf(m19a, fminf(fabsf(c0 - tile[b0 + o0]),      fabsf(c0 - tile[b0 + o1])));
    m19b = fminf(m19b, fminf(fabsf(c1 - tile[b0 + RB + o0]), fabsf(c1 - tile[b0 + RB + o1])));
  }

  const float w1 = *P1, w2 = *P2, w3 = *P3, w4 = *P4;
  const float inv = 1.0f / (w1 + w2 + w3 + w4);

  // max_i (1 - d_i) == 1 - min_i d_i  (monotone rounding -> bit-identical)
  const float o0v = (w1 * (1.0f - m7a) + w2 * (1.0f - m11a) +
                     w3 * (1.0f - m15a) + w4 * (1.0f - m19a)) * inv;
  const float o1v = (w1 * (1.0f - m7b) + w2 * (1.0f - m11b) +
                     w3 * (1.0f - m15b) + w4 * (1.0f - m19b)) * inv;

  const int gx = ox + tx;
  dst[(oy + tw)      * LIMG + gx] = o0v;
  dst[(oy + tw + 16) * LIMG + gx] = o1v;
}

extern "C" void kernel_launch(void* const* d_in, const int* in_sizes, int n_in,
                              void* d_out, int out_size, void* d_ws, size_t ws_size,
                              hipStream_t stream) {
  const float* x  = (const float*)d_in[0];
  const float* p1 = (const float*)d_in[1];
  const float* p2 = (const float*)d_in[2];
  const float* p3 = (const float*)d_in[3];
  const float* p4 = (const float*)d_in[4];
  float* out = (float*)d_out;
  float* tmp = (float*)d_ws;          // 4 MB intermediate (layer-1 output)

  dim3 grid(LIMG / TILE, LIMG / TILE);
  geneo_layer<<<grid, NTHR, 0, stream>>>(x,   tmp, p1, p2, p3, p4);   // L1
  geneo_layer<<<grid, NTHR, 0, stream>>>(tmp, out, p1, p2, p3, p4);   // L2
}